// _SolarOpenMoE_35691178230203
// MI455X (gfx1250) — compile-verified
//
#include <hip/hip_runtime.h>
#include <hip/hip_bf16.h>

// ---------------------------------------------------------------------------
// Problem constants (match reference)
// ---------------------------------------------------------------------------
constexpr int kH   = 1024;   // hidden
constexpr int kI   = 512;    // moe intermediate
constexpr int kE   = 32;     // routed experts
constexpr int kTopK   = 6;
constexpr int kNGroup = 8;
constexpr int kGSize  = kE / kNGroup;   // 4
constexpr int kTopKG  = 4;
constexpr int kSI  = 1024;   // shared intermediate
constexpr int kT   = 2048;   // tokens (B*S)
constexpr float kScale = 2.5f;

// LDS row strides padded so stride % 64banks = 4: a wave32 b64 A-load
// (lanes m=0..15, koff in {0,2}) maps to banks 4m+2h+{0,1} = all 64 banks
// exactly once -> conflict-free.
constexpr int kXPad  = kH + 4;   // 1028 floats/row for the x tile
constexpr int kSPad  = 512 + 4;  // 516 floats/row for the intermediate tile

typedef float v2f __attribute__((ext_vector_type(2)));
typedef float v8f __attribute__((ext_vector_type(8)));

// D = A(16x4 f32) x B(4x16 f32) + C, full fp32 accumulate (matches reference
// numerics). Per ISA 7.12.2: A lane m=lane&15 holds K = koff,koff+1 in its two
// VGPRs (koff = 2*(lane>=16)); B mirrors with lane n=lane&15; C/D v8f with
// row m = v + 8*(lane>=16), col n = lane&15.
__device__ __forceinline__ v8f wmma4(v2f a, v2f b, v8f c) {
  return __builtin_amdgcn_wmma_f32_16x16x4_f32(
      /*neg_a=*/false, a, /*neg_b=*/false, b,
      /*c_mod=*/(short)0, c, /*reuse_a=*/false, /*reuse_b=*/false);
}

__device__ __forceinline__ v8f vzero8() {
  v8f z = {0.f, 0.f, 0.f, 0.f, 0.f, 0.f, 0.f, 0.f};
  return z;
}

// sigmoid via hardware v_rcp_f32 (~1 ulp) instead of the IEEE div expansion.
__device__ __forceinline__ float fast_sigmoid(float g) {
  return __builtin_amdgcn_rcpf(1.0f + __expf(-g));
}
__device__ __forceinline__ float silu(float g) {
  return g * fast_sigmoid(g);
}

// Cooperative coalesced load of a 16 x kH f32 tile into LDS (padded rows).
// 256 threads: 16 threads per row, each copies 16 float4 chunks.
__device__ __forceinline__ void load_xtile(float* __restrict__ xt,
                                           const float* __restrict__ x,
                                           const int* __restrict__ rows,
                                           int tid) {
  const int r = tid >> 4;          // row 0..15
  const int j = tid & 15;          // 16 threads per row
  const float* src = x + (size_t)rows[r] * kH;
  float* dst = xt + r * kXPad;
  #pragma unroll
  for (int c = 0; c < 16; ++c) {
    const int col = (j + c * 16) * 4;     // float index, 16B aligned
    *(float4*)(dst + col) = *(const float4*)(src + col);
  }
}

// ---------------------------------------------------------------------------
// Kernel 1: router. One wave32 per token; lane e computes logit for expert e.
// Lane 0 then does the (tiny) group-limited top-k serially and scatters
// (token, combine-weight) into per-expert gather lists.
// ---------------------------------------------------------------------------
__global__ __launch_bounds__(128)
void router_kernel(const float* __restrict__ x,
                   const float* __restrict__ gate_w,     // [E,H]
                   const float* __restrict__ bias,       // [E]
                   int*   __restrict__ cnt,              // [E], pre-zeroed
                   int*   __restrict__ list_tok,         // [E,T]
                   float* __restrict__ list_w)           // [E,T]
{
  __shared__ float s_sc[4][kE];
  __shared__ float s_sfc[4][kE];
  const int wave = threadIdx.x >> 5;
  const int lane = threadIdx.x & 31;
  const int t = blockIdx.x * 4 + wave;

  // lane e: logit = dot(x[t], gate_w[e])
  const float* xr = x + (size_t)t * kH;
  const float* wr = gate_w + (size_t)lane * kH;
  float acc = 0.f;
  for (int h = 0; h < kH; h += 4) {
    acc += xr[h + 0] * wr[h + 0];
    acc += xr[h + 1] * wr[h + 1];
    acc += xr[h + 2] * wr[h + 2];
    acc += xr[h + 3] * wr[h + 3];
  }
  const float sc = fast_sigmoid(acc);
  s_sc[wave][lane]  = sc;
  s_sfc[wave][lane] = sc + bias[lane];
  __syncthreads();

  if (lane == 0) {
    const float* sfc = s_sfc[wave];
    const float* ssc = s_sc[wave];
    // group scores = max within group of 4
    float gsc[kNGroup];
    for (int g = 0; g < kNGroup; ++g) {
      float m = sfc[g * kGSize];
      for (int j = 1; j < kGSize; ++j) m = fmaxf(m, sfc[g * kGSize + j]);
      gsc[g] = m;
    }
    // top-4 groups (strict > keeps lowest index on tie, like jax top_k)
    bool gsel[kNGroup] = {};
    for (int it = 0; it < kTopKG; ++it) {
      int best = -1; float bv = -1e30f;
      for (int g = 0; g < kNGroup; ++g)
        if (!gsel[g] && gsc[g] > bv) { bv = gsc[g]; best = g; }
      gsel[best] = true;
    }
    // masked scores, then top-6 experts
    float m2[kE];
    for (int e = 0; e < kE; ++e) m2[e] = gsel[e / kGSize] ? sfc[e] : 0.0f;
    int   idx[kTopK];
    float wv[kTopK];
    float wsum = 0.f;
    for (int j = 0; j < kTopK; ++j) {
      int best = 0; float bv = -1e30f;
      for (int e = 0; e < kE; ++e)
        if (m2[e] > bv) { bv = m2[e]; best = e; }
      m2[best] = -1e30f;
      idx[j] = best;
      wv[j]  = ssc[best];          // weight from raw sigmoid score
      wsum  += wv[j];
    }
    const float inv = kScale / (wsum + 1e-20f);
    for (int j = 0; j < kTopK; ++j) {
      const int e = idx[j];
      const int pos = atomicAdd(&cnt[e], 1);
      list_tok[e * kT + pos] = t;
      list_w[e * kT + pos]   = wv[j] * inv;
    }
  }
}

// ---------------------------------------------------------------------------
// Kernel 2: shared experts. 16-token tile per block, 8 waves.
// x tile staged once in LDS. Stage A: one LDS A-load feeds 8 WMMAs
// (4 column tiles x gate/up) per K-step; sh chunk -> padded LDS.
// Stage B: 64 persistent accumulator VGPRs/wave over both 512-wide chunks;
// plain stores fully cover out so the routed kernel can atomic-add on top.
// ---------------------------------------------------------------------------
__global__ __launch_bounds__(256)
void shared_kernel(const float* __restrict__ x,
                   const float* __restrict__ gw,   // [SI,H]
                   const float* __restrict__ uw,   // [SI,H]
                   const float* __restrict__ dw,   // [H,SI]
                   float* __restrict__ out)        // [T,H]
{
  __shared__ float xt[16 * kXPad];                 // ~64KB x tile
  __shared__ float sh[16 * kSPad];                 // ~33KB intermediate chunk
  __shared__ int   s_rows[16];

  const int tid  = threadIdx.x;
  const int wave = tid >> 5;
  const int lane = tid & 31;
  const int mrow = lane & 15;
  const int koff = (lane >> 4) << 1;
  const int rbase = (lane >> 4) * 8;
  const int t0 = blockIdx.x * 16;

  if (tid < 16) s_rows[tid] = t0 + tid;
  __syncthreads();
  load_xtile(xt, x, s_rows, tid);
  __syncthreads();

  const float* arow = xt + mrow * kXPad + koff;    // LDS A stream

  v8f acc[8];
  #pragma unroll
  for (int tI = 0; tI < 8; ++tI) acc[tI] = vzero8();

  for (int c = 0; c < 2; ++c) {
    if (c) __syncthreads();                        // stage B of prev chunk done
    // ---- stage A: this wave computes 64 of this chunk's 512 columns
    v8f ga[4], ua[4];
    const float* grow[4];
    const float* urow[4];
    #pragma unroll
    for (int ct = 0; ct < 4; ++ct) {
      ga[ct] = vzero8(); ua[ct] = vzero8();
      const int n = c * 512 + wave * 64 + ct * 16 + mrow;  // global SI col
      grow[ct] = gw + (size_t)n * kH + koff;
      urow[ct] = uw + (size_t)n * kH + koff;
    }
    for (int k = 0; k < kH; k += 4) {
      v2f a = *(const v2f*)(arow + k);             // 1 ds load -> 8 wmma
      #pragma unroll
      for (int ct = 0; ct < 4; ++ct) {
        v2f bg = *(const v2f*)(grow[ct] + k);
        ga[ct] = wmma4(a, bg, ga[ct]);
        v2f bu = *(const v2f*)(urow[ct] + k);
        ua[ct] = wmma4(a, bu, ua[ct]);
      }
    }
    #pragma unroll
    for (int ct = 0; ct < 4; ++ct) {
      const int nloc = wave * 64 + ct * 16 + mrow;
      #pragma unroll
      for (int v = 0; v < 8; ++v)
        sh[(rbase + v) * kSPad + nloc] = silu(ga[ct][v]) * ua[ct][v];
    }
    __syncthreads();
    // ---- stage B: accumulate down-proj for this chunk's K range
    for (int k = 0; k < 512; k += 4) {
      v2f a = *(const v2f*)(&sh[mrow * kSPad + koff + k]);  // reused 8x
      #pragma unroll
      for (int tI = 0; tI < 8; ++tI) {
        const int n = (wave * 8 + tI) * 16 + mrow;          // out col (H)
        v2f b = *(const v2f*)(dw + (size_t)n * kSI + c * 512 + koff + k);
        acc[tI] = wmma4(a, b, acc[tI]);
      }
    }
  }

  #pragma unroll
  for (int tI = 0; tI < 8; ++tI) {
    const int n = (wave * 8 + tI) * 16 + mrow;
    #pragma unroll
    for (int v = 0; v < 8; ++v)
      out[(size_t)(t0 + rbase + v) * kH + n] = acc[tI][v];
  }
}

// ---------------------------------------------------------------------------
// Kernel 3: routed experts over gathered token lists. grid = (128 tiles, 32
// experts); blocks past an expert's token count exit immediately (uniform).
// Same two-stage structure as the shared kernel, with gathered x rows and
// combine-weight scaling; results accumulated via global_atomic_add_f32.
// ---------------------------------------------------------------------------
__global__ __launch_bounds__(256)
void routed_kernel(const float* __restrict__ x,
                   const float* __restrict__ egu,   // [E,2I,H]
                   const float* __restrict__ edn,   // [E,H,I]
                   const int*   __restrict__ cnt,   // [E]
                   const int*   __restrict__ list_tok,
                   const float* __restrict__ list_w,
                   float* __restrict__ out)         // [T,H]
{
  const int e    = blockIdx.y;
  const int tile = blockIdx.x;
  const int nt   = cnt[e];
  if (tile * 16 >= nt) return;                      // uniform early-out

  __shared__ float xt[16 * kXPad];                  // ~64KB gathered x tile
  __shared__ float sh[16 * kSPad];                  // ~33KB intermediate
  __shared__ int   s_tok[16];
  __shared__ float s_cw[16];

  const int tid  = threadIdx.x;
  if (tid < 16) {
    const int m = tile * 16 + tid;
    if (m < nt) { s_tok[tid] = list_tok[e * kT + m]; s_cw[tid] = list_w[e * kT + m]; }
    else        { s_tok[tid] = 0;                    s_cw[tid] = 0.f; }
  }
  __syncthreads();
  load_xtile(xt, x, s_tok, tid);
  __syncthreads();

  const int wave = tid >> 5;
  const int lane = tid & 31;
  const int mrow = lane & 15;
  const int koff = (lane >> 4) << 1;
  const int rbase = (lane >> 4) * 8;

  const float* arow  = xt + mrow * kXPad + koff;    // LDS A stream
  const float* gbase = egu + (size_t)e * (2 * kI) * kH;

  // ---- stage A: 4 column tiles x (gate, up) share each A load
  v8f ga[4], ua[4];
  const float* grow[4];
  const float* urow[4];
  #pragma unroll
  for (int ct = 0; ct < 4; ++ct) {
    ga[ct] = vzero8(); ua[ct] = vzero8();
    const int n = wave * 64 + ct * 16 + mrow;       // intermediate col
    grow[ct] = gbase + (size_t)n * kH + koff;
    urow[ct] = gbase + (size_t)(kI + n) * kH + koff;
  }
  for (int k = 0; k < kH; k += 4) {
    v2f a = *(const v2f*)(arow + k);
    #pragma unroll
    for (int ct = 0; ct < 4; ++ct) {
      v2f bg = *(const v2f*)(grow[ct] + k);
      ga[ct] = wmma4(a, bg, ga[ct]);
      v2f bu = *(const v2f*)(urow[ct] + k);
      ua[ct] = wmma4(a, bu, ua[ct]);
    }
  }
  #pragma unroll
  for (int ct = 0; ct < 4; ++ct) {
    const int nloc = wave * 64 + ct * 16 + mrow;
    #pragma unroll
    for (int v = 0; v < 8; ++v)
      sh[(rbase + v) * kSPad + nloc] = silu(ga[ct][v]) * ua[ct][v];
  }
  __syncthreads();

  // ---- stage B
  const float* dbase = edn + (size_t)e * kH * kI;
  v8f acc[8];
  #pragma unroll
  for (int tI = 0; tI < 8; ++tI) acc[tI] = vzero8();
  for (int k = 0; k < kI; k += 4) {
    v2f a = *(const v2f*)(&sh[mrow * kSPad + koff + k]);
    #pragma unroll
    for (int tI = 0; tI < 8; ++tI) {
      const int n = (wave * 8 + tI) * 16 + mrow;    // out col (H)
      v2f b = *(const v2f*)(dbase + (size_t)n * kI + koff + k);
      acc[tI] = wmma4(a, b, acc[tI]);
    }
  }

  #pragma unroll
  for (int tI = 0; tI < 8; ++tI) {
    const int n = (wave * 8 + tI) * 16 + mrow;
    #pragma unroll
    for (int v = 0; v < 8; ++v) {
      const int m = rbase + v;
      atomicAdd(&out[(size_t)s_tok[m] * kH + n], acc[tI][v] * s_cw[m]);
    }
  }
}

// ---------------------------------------------------------------------------
// Launch
// ---------------------------------------------------------------------------
extern "C" void kernel_launch(void* const* d_in, const int* in_sizes, int n_in,
                              void* d_out, int out_size, void* d_ws, size_t ws_size,
                              hipStream_t stream) {
  const float* x    = (const float*)d_in[0];
  const float* gw   = (const float*)d_in[1];
  const float* bias = (const float*)d_in[2];
  const float* egu  = (const float*)d_in[3];
  const float* edn  = (const float*)d_in[4];
  const float* sgw  = (const float*)d_in[5];
  const float* suw  = (const float*)d_in[6];
  const float* sdw  = (const float*)d_in[7];
  float* out = (float*)d_out;

  char* ws = (char*)d_ws;
  int*   cnt      = (int*)ws;                                    // E ints
  int*   list_tok = (int*)(ws + 256);                            // E*T ints
  float* list_w   = (float*)(ws + 256 + (size_t)kE * kT * 4);    // E*T floats

  hipMemsetAsync(cnt, 0, kE * sizeof(int), stream);
  router_kernel<<<kT / 4, 128, 0, stream>>>(x, gw, bias, cnt, list_tok, list_w);
  // shared expert fully overwrites out (plain stores), then routed adds.
  shared_kernel<<<kT / 16, 256, 0, stream>>>(x, sgw, suw, sdw, out);
  dim3 grid(kT / 16, kE);
  routed_kernel<<<grid, 256, 0, stream>>>(x, egu, edn, cnt, list_tok, list_w, out);
}